// PrototypeStats_57836029608240
// MI455X (gfx1250) — compile-verified
//
#include <hip/hip_runtime.h>
#include <hip/hip_bf16.h>

typedef __attribute__((ext_vector_type(2))) float v2f;
typedef __attribute__((ext_vector_type(8))) float v8f;

#define NB 8
#define NN 8192
#define ND 1024
#define NK 64
#define TEMP_INV 10.0f
#define EPSF 1e-6f
#define TILE_N 128
#define DC 32        // d-chunk per pipeline stage
#define NCHUNK (ND / DC)
#define DPZ 36       // padded stride for z/P tiles (144B rows: 16B aligned, banks 36m%64 distinct)
#define DPL 68       // padded stride for logits/assign (64-wide rows)

// output offsets (floats) in d_out: assign | count | mean | var
#define OFF_ASSIGN 0u
#define OFF_COUNT  4194304u
#define OFF_MEAN   4194816u
#define OFF_VAR    4719104u

__device__ __forceinline__ v8f wmma_f32(v2f a, v2f b, v8f c) {
  // D = A(16x4 f32) * B(4x16 f32) + C(16x16 f32)
  return __builtin_amdgcn_wmma_f32_16x16x4_f32(false, a, false, b, (short)0, c,
                                               false, false);
}

// CDNA5 async global->LDS copy, tracked by ASYNCcnt (no VGPR round trip).
// INST_OFFSET applies to BOTH the global and LDS addresses (ISA 15.18.3), which
// matches our equal 16B strides on both sides.
__device__ __forceinline__ void async_b128(unsigned lds, const float* g, int imm) {
  switch (imm) {
    case 0:  asm volatile("global_load_async_to_lds_b128 %0, %1, off"           :: "v"(lds), "v"(g) : "memory"); break;
    case 16: asm volatile("global_load_async_to_lds_b128 %0, %1, off offset:16" :: "v"(lds), "v"(g) : "memory"); break;
    case 32: asm volatile("global_load_async_to_lds_b128 %0, %1, off offset:32" :: "v"(lds), "v"(g) : "memory"); break;
    default: asm volatile("global_load_async_to_lds_b128 %0, %1, off offset:48" :: "v"(lds), "v"(g) : "memory"); break;
  }
}
#define WAIT_ASYNC_6() asm volatile("s_wait_asynccnt 0x6" ::: "memory")
#define WAIT_ASYNC_4() asm volatile("s_wait_asynccnt 0x4" ::: "memory")
#define WAIT_ASYNC_0() asm volatile("s_wait_asynccnt 0x0" ::: "memory")

__device__ __forceinline__ unsigned lds_off(const void* p) {
  return (unsigned)(size_t)p;   // generic shared pointer: low 32 bits = LDS offset
}

extern "C" __global__ void __launch_bounds__(256)
k_proto_main(const float* __restrict__ z, const float* __restrict__ P,
             float* __restrict__ assign_out,
             float* __restrict__ cnt_raw, float* __restrict__ al_raw,
             float* __restrict__ az2_raw, float* __restrict__ mean_raw)
{
  __shared__ float zt[2][TILE_N][DPZ];  // double-buffered z tile (rows x DC)
  __shared__ float pt[2][NK][DPZ];      // double-buffered prototype tile (K x DC)
  __shared__ float lg[TILE_N][DPL];     // logits (rows x 64)
  __shared__ float as_[TILE_N][DPL];    // assign (rows x 64)
  __shared__ float z2p[TILE_N][2];      // per-row ||z||^2 half-partials
  __shared__ float z2s[TILE_N];         // per-row ||z||^2
  __shared__ float red[3][4][NK];       // block partials: count, a*logit, a*z2

  const int t    = threadIdx.x;
  const int lane = t & 31;
  const int wave = t >> 5;
  const int c15  = lane & 15;
  const int hi   = lane >> 4;
  const int blk  = blockIdx.x;          // 512 = 8 batches * 64 row tiles
  const int b    = blk >> 6;
  const int n0   = (blk & 63) * TILE_N;
  const float* zb = z + ((size_t)b * NN + n0) * ND;

  // async staging maps (uniform 6 loads/thread/chunk in phase1; 4 in phase2)
  const int zrow = t >> 1;              // 2 threads per z row, 16 floats each
  const int zcol = (t & 1) * 16;
  const int prow = t >> 2;              // 4 threads per P row, 8 floats each
  const int pcol = (t & 3) * 8;

  // ---------------- Phase 1: logits = z . P^T (pipelined async -> WMMA) ----
  v8f acc0 = {}, acc1 = {}, acc2 = {}, acc3 = {};
  float sumsq = 0.0f;

  // prologue: stage chunk 0 into buffer 0
  {
    const float* gz = zb + (size_t)zrow * ND + zcol;
    unsigned lz = lds_off(&zt[0][zrow][zcol]);
    async_b128(lz, gz, 0); async_b128(lz, gz, 16);
    async_b128(lz, gz, 32); async_b128(lz, gz, 48);
    const float* gp = P + (size_t)prow * ND + pcol;
    unsigned lp = lds_off(&pt[0][prow][pcol]);
    async_b128(lp, gp, 0); async_b128(lp, gp, 16);
  }

  for (int c = 0; c < NCHUNK; ++c) {
    const int buf = c & 1;
    __syncthreads();                    // everyone done reading buf^1
    if (c + 1 < NCHUNK) {
      const int dc = (c + 1) * DC;
      const float* gz = zb + (size_t)zrow * ND + dc + zcol;
      unsigned lz = lds_off(&zt[buf ^ 1][zrow][zcol]);
      async_b128(lz, gz, 0); async_b128(lz, gz, 16);
      async_b128(lz, gz, 32); async_b128(lz, gz, 48);
      const float* gp = P + (size_t)prow * ND + dc + pcol;
      unsigned lp = lds_off(&pt[buf ^ 1][prow][pcol]);
      async_b128(lp, gp, 0); async_b128(lp, gp, 16);
      WAIT_ASYNC_6();                   // retire chunk c, keep c+1 in flight
    } else {
      WAIT_ASYNC_0();
    }
    __syncthreads();                    // chunk c visible to all waves

    const int m    = wave * 16 + c15;
    const int koff = hi * 2;
    #pragma unroll
    for (int dk = 0; dk < DC; dk += 4) {
      v2f a  = *reinterpret_cast<const v2f*>(&zt[buf][m][dk + koff]);
      sumsq += a.x * a.x + a.y * a.y;   // fold ||z||^2 into A-fragment loads
      v2f b0 = *reinterpret_cast<const v2f*>(&pt[buf][ 0 + c15][dk + koff]);
      v2f b1 = *reinterpret_cast<const v2f*>(&pt[buf][16 + c15][dk + koff]);
      v2f b2 = *reinterpret_cast<const v2f*>(&pt[buf][32 + c15][dk + koff]);
      v2f b3 = *reinterpret_cast<const v2f*>(&pt[buf][48 + c15][dk + koff]);
      acc0 = wmma_f32(a, b0, acc0);
      acc1 = wmma_f32(a, b1, acc1);
      acc2 = wmma_f32(a, b2, acc2);
      acc3 = wmma_f32(a, b3, acc3);
    }
  }
  // lane (c15,hi) saw exactly the {2*hi, 2*hi+1} mod 4 elements of row m
  z2p[wave * 16 + c15][hi] = sumsq;
  __syncthreads();

  // dump logits to LDS (C layout: VGPR v -> M = v + 8*hi, N = c15)
  {
    const int rbase = wave * 16 + hi * 8;
    #pragma unroll
    for (int v = 0; v < 8; ++v) {
      lg[rbase + v][ 0 + c15] = acc0[v];
      lg[rbase + v][16 + c15] = acc1[v];
      lg[rbase + v][32 + c15] = acc2[v];
      lg[rbase + v][48 + c15] = acc3[v];
    }
  }
  __syncthreads();

  // ---------------- softmax over K=64, one thread per row ----------------
  if (t < TILE_N) {
    float mx = -3.4e38f;
    #pragma unroll 8
    for (int k = 0; k < NK; ++k) mx = fmaxf(mx, lg[t][k]);
    float s = 0.0f;
    #pragma unroll 8
    for (int k = 0; k < NK; ++k) {
      float e = __expf((lg[t][k] - mx) * TEMP_INV);
      as_[t][k] = e;
      s += e;
    }
    float inv = 1.0f / s;
    #pragma unroll 8
    for (int k = 0; k < NK; ++k) as_[t][k] *= inv;
    z2s[t] = z2p[t][0] + z2p[t][1];
  }
  __syncthreads();

  // write assign out (coalesced float4)
  {
    const int arow = t >> 1, acol = (t & 1) * 32;
    float* aout = assign_out + ((size_t)b * NN + n0 + arow) * NK + acol;
    const float* asrc = &as_[arow][acol];
    #pragma unroll
    for (int j = 0; j < 8; ++j)
      reinterpret_cast<float4*>(aout)[j] = *reinterpret_cast<const float4*>(asrc + 4 * j);
  }

  // block partials: count[k], sum a*logit, sum a*z2
  {
    const int k = t & 63;
    const int g = t >> 6;
    float c = 0.0f, la = 0.0f, a2 = 0.0f;
    #pragma unroll 4
    for (int r = g * 32; r < g * 32 + 32; ++r) {
      float a = as_[r][k];
      c  += a;
      la += a * lg[r][k];
      a2 += a * z2s[r];
    }
    red[0][g][k] = c; red[1][g][k] = la; red[2][g][k] = a2;
  }
  __syncthreads();
  if (t < NK) {
    float c  = red[0][0][t] + red[0][1][t] + red[0][2][t] + red[0][3][t];
    float la = red[1][0][t] + red[1][1][t] + red[1][2][t] + red[1][3][t];
    float a2 = red[2][0][t] + red[2][1][t] + red[2][2][t] + red[2][3][t];
    atomicAdd(&cnt_raw[b * NK + t], c);
    atomicAdd(&al_raw [b * NK + t], la);
    atomicAdd(&az2_raw[b * NK + t], a2);
  }

  // ---------------- Phase 2: mean_raw += assign^T . z (pipelined) ---------
  const int mt = wave & 3;              // M tile (16 protos)
  const int nt = wave >> 2;             // N tile (16 of the 32 d-columns)
  float* mb = mean_raw + (size_t)b * NK * ND;

  // prologue: stage chunk 0 z into buffer 0
  {
    const float* gz = zb + (size_t)zrow * ND + zcol;
    unsigned lz = lds_off(&zt[0][zrow][zcol]);
    async_b128(lz, gz, 0); async_b128(lz, gz, 16);
    async_b128(lz, gz, 32); async_b128(lz, gz, 48);
  }
  for (int c = 0; c < NCHUNK; ++c) {
    const int buf = c & 1;
    const int dc  = c * DC;
    __syncthreads();
    if (c + 1 < NCHUNK) {
      const float* gz = zb + (size_t)zrow * ND + (dc + DC) + zcol;
      unsigned lz = lds_off(&zt[buf ^ 1][zrow][zcol]);
      async_b128(lz, gz, 0); async_b128(lz, gz, 16);
      async_b128(lz, gz, 32); async_b128(lz, gz, 48);
      WAIT_ASYNC_4();
    } else {
      WAIT_ASYNC_0();
    }
    __syncthreads();

    v8f m0 = {};
    const int dcol = nt * 16 + c15;
    const int kp   = mt * 16 + c15;
    const int roff = hi * 2;
    #pragma unroll 8
    for (int r = 0; r < TILE_N; r += 4) {
      v2f bf = { zt[buf][r + roff][dcol], zt[buf][r + roff + 1][dcol] };  // B[row][d]
      v2f af = { as_[r + roff][kp],       as_[r + roff + 1][kp] };        // A[k][row]
      m0 = wmma_f32(af, bf, m0);
    }
    {
      const int dd = dc + nt * 16 + c15;
      const int mm = mt * 16 + hi * 8;
      #pragma unroll
      for (int v = 0; v < 8; ++v)
        atomicAdd(&mb[(size_t)(mm + v) * ND + dd], m0[v]);
    }
  }
}

// p2[k] = sum_d P[k,d]^2  (one block)
extern "C" __global__ void __launch_bounds__(256)
k_proto_p2(const float* __restrict__ P, float* __restrict__ p2)
{
  __shared__ float red[256];
  const int t = threadIdx.x, k = t >> 2, q = t & 3;
  const float* row = P + (size_t)k * ND + q * 256;
  float s = 0.0f;
  #pragma unroll 8
  for (int j = 0; j < 256; j += 4) {
    float4 v = *reinterpret_cast<const float4*>(row + j);
    s += v.x * v.x + v.y * v.y + v.z * v.z + v.w * v.w;
  }
  red[t] = s;
  __syncthreads();
  if (q == 0) p2[k] = red[t] + red[t + 1] + red[t + 2] + red[t + 3];
}

extern "C" __global__ void __launch_bounds__(256)
k_proto_fin_mean(const float* __restrict__ mean_raw, const float* __restrict__ cnt_raw,
                 float* __restrict__ mean_out)
{
  const int i = blockIdx.x * 256 + threadIdx.x;   // 8*64*1024 total
  const int bk = i >> 10;
  mean_out[i] = mean_raw[i] / (cnt_raw[bk] + EPSF);
}

extern "C" __global__ void __launch_bounds__(256)
k_proto_fin_small(const float* __restrict__ cnt_raw, const float* __restrict__ al_raw,
                  const float* __restrict__ az2_raw, const float* __restrict__ p2,
                  float* __restrict__ count_out, float* __restrict__ var_out)
{
  const int i = blockIdx.x * 256 + threadIdx.x;
  if (i < NB * NK) {
    const int k = i & 63;
    float c  = cnt_raw[i];
    float cn = c + EPSF;
    count_out[i] = cn;
    // sum_n a*(z2 + p2 - 2*logit) = az2 + p2*count - 2*al
    var_out[i] = (az2_raw[i] + p2[k] * c - 2.0f * al_raw[i]) / cn;
  }
}

extern "C" void kernel_launch(void* const* d_in, const int* in_sizes, int n_in,
                              void* d_out, int out_size, void* d_ws, size_t ws_size,
                              hipStream_t stream) {
  const float* z = (const float*)d_in[0];
  const float* P = (const float*)d_in[1];
  float* out        = (float*)d_out;
  float* assign_out = out + OFF_ASSIGN;
  float* count_out  = out + OFF_COUNT;
  float* mean_out   = out + OFF_MEAN;
  float* var_out    = out + OFF_VAR;

  float* ws       = (float*)d_ws;
  float* cnt_raw  = ws;               // 512
  float* al_raw   = ws + 512;         // 512
  float* az2_raw  = ws + 1024;        // 512
  float* p2       = ws + 1536;        // 64 (+ pad to 2048)
  float* mean_raw = ws + 2048;        // 8*64*1024 = 524288

  hipMemsetAsync(d_ws, 0, (size_t)(2048 + NB * NK * ND) * sizeof(float), stream);
  k_proto_p2<<<1, 256, 0, stream>>>(P, p2);
  k_proto_main<<<NB * (NN / TILE_N), 256, 0, stream>>>(z, P, assign_out,
                                                       cnt_raw, al_raw, az2_raw, mean_raw);
  k_proto_fin_mean<<<(NB * NK * ND) / 256, 256, 0, stream>>>(mean_raw, cnt_raw, mean_out);
  k_proto_fin_small<<<2, 256, 0, stream>>>(cnt_raw, al_raw, az2_raw, p2, count_out, var_out);
}